// RNNLoop_12489764897121
// MI455X (gfx1250) — compile-verified
//
#include <hip/hip_runtime.h>

typedef __attribute__((ext_vector_type(2))) float v2f;
typedef __attribute__((ext_vector_type(8))) float v8f;

static constexpr int LL = 3;     // layers
static constexpr int DD = 10;    // feature dim
static constexpr int BB = 4096;  // batch
static constexpr int TT = 2048;  // time steps
static constexpr int CS = 28;    // LDS cat row stride (floats): even (8B align),
                                 // >25 (pad-lane inp stores land in dead cols),
                                 // 28m mod 64 hits 16 distinct banks, 2addr offsets < 256

// One wave32 owns a 16-row batch tile. Layer states live in registers in the
// WMMA C/D layout (lane = feature column n, VGPR j = batch row j + 8*half).
// Each layer update D = A(16x20 cat) x B(20x16 = W^T, cols 10..15 zero) + C
// is 5 chained V_WMMA_F32_16X16X4_F32. The D->A transpose goes through a
// per-wave LDS tile. All 32 lanes store unconditionally (pad lanes n>=10 write
// zeros into cols 10..15, later overwritten by the in-order inp store, or into
// dead cols 20..25) -- no exec-mask branching in the hot loop.
__global__ __launch_bounds__(32) void rnn_wmma_kernel(
    const float* __restrict__ initial,  // [3, 4096, 10]
    const float* __restrict__ xseq,     // [4096, 2048, 10]
    const float* __restrict__ Ws,       // [3, 10, 20]
    const float* __restrict__ bs,       // [3, 10]
    float* __restrict__ out)            // [4096, 2048, 10]
{
    __shared__ float cat[16][CS];  // rows = batch row in tile; cols 0..9 = h, 10..19 = inp

    const int lane = threadIdx.x & 31;
    const int n    = lane & 15;   // B/C/D column (feature) for this lane
    const int hh   = lane >> 4;   // half-wave select (K split for A/B tiles)
    const int m    = n;           // A-matrix row index = lane & 15
    const int rowBase = blockIdx.x * 16;

    // ---- Preload B-matrix tiles: B[k][n] = W[n][k], zero for n >= 10 ----
    // Tile kt: .x = B[4kt + 2*hh][n], .y = B[4kt + 2*hh + 1][n]
    v2f  wb[LL][5];
    float bias[LL];
    #pragma unroll
    for (int li = 0; li < LL; ++li) {
        const float* W = Ws + li * DD * 2 * DD;
        #pragma unroll
        for (int kt = 0; kt < 5; ++kt) {
            const int k0 = 4 * kt + 2 * hh;
            v2f w;
            w.x = (n < DD) ? W[n * 2 * DD + k0]     : 0.0f;
            w.y = (n < DD) ? W[n * 2 * DD + k0 + 1] : 0.0f;
            wb[li][kt] = w;
        }
        bias[li] = (n < DD) ? bs[li * DD + n] : 0.0f;
    }

    // ---- Load initial state in C/D layout; pad columns n>=10 stay exactly 0 ----
    v8f h[LL];
    #pragma unroll
    for (int li = 0; li < LL; ++li) {
        v8f s = {};
        #pragma unroll
        for (int j = 0; j < 8; ++j) {
            const int row = rowBase + j + 8 * hh;
            s[j] = (n < DD) ? initial[(size_t)li * BB * DD + (size_t)row * DD + n] : 0.0f;
        }
        h[li] = s;
    }

    const float* xrow    = xseq + (size_t)(rowBase + m) * TT * DD;           // A-row source
    float*       outLane = out  + (size_t)(rowBase + 8 * hh) * TT * DD + n;  // + (j*T + t)*D

    #pragma unroll 1
    for (int t = 0; t < TT; ++t) {
        const float* xt = xrow + (size_t)t * DD;
        __builtin_prefetch(xt + DD, 0, 0);  // next step's x row -> global_prefetch_b8
        v8f inpState = {};

        #pragma unroll
        for (int li = 0; li < LL; ++li) {
            // Stage h_old transposed into cols 0..15 (pad lanes write zeros to 10..15)
            #pragma unroll
            for (int j = 0; j < 8; ++j) cat[j + 8 * hh][n] = h[li][j];
            if (li > 0) {
                // Must execute after the h store (in-order DS within the wave) so
                // real inp values overwrite the pad-lane zeros at cols 10..15.
                asm volatile("" ::: "memory");
                #pragma unroll
                for (int j = 0; j < 8; ++j) cat[j + 8 * hh][DD + n] = inpState[j];
            }
            asm volatile("" ::: "memory");  // keep DS stores before DS loads

            // Build A tiles: lane holds A[m][k0], A[m][k0+1] with k0 = 4*kt + 2*hh
            v2f a0, a1, a2, a3, a4;
            a0 = *(const v2f*)&cat[m][0 + 2 * hh];
            a1 = *(const v2f*)&cat[m][4 + 2 * hh];
            if (li == 0) {
                // tile 2 straddles h (cols 8,9 -> half 0) and x (cols 0,1 -> half 1)
                v2f a2l = *(const v2f*)&cat[m][8 + 2 * hh];
                v2f a2g; a2g.x = xt[0]; a2g.y = xt[1];
                a2 = hh ? a2g : a2l;
                a3.x = xt[2 + 2 * hh]; a3.y = xt[3 + 2 * hh];
                a4.x = xt[6 + 2 * hh]; a4.y = xt[7 + 2 * hh];
            } else {
                a2 = *(const v2f*)&cat[m][8 + 2 * hh];
                a3 = *(const v2f*)&cat[m][12 + 2 * hh];
                a4 = *(const v2f*)&cat[m][16 + 2 * hh];
            }

            // C = h_old + bias (residual + bias folded into the accumulator)
            v8f acc = h[li];
            #pragma unroll
            for (int j = 0; j < 8; ++j) acc[j] += bias[li];

            acc = __builtin_amdgcn_wmma_f32_16x16x4_f32(false, a0, false, wb[li][0], (short)0, acc, false, false);
            acc = __builtin_amdgcn_wmma_f32_16x16x4_f32(false, a1, false, wb[li][1], (short)0, acc, false, false);
            acc = __builtin_amdgcn_wmma_f32_16x16x4_f32(false, a2, false, wb[li][2], (short)0, acc, false, false);
            acc = __builtin_amdgcn_wmma_f32_16x16x4_f32(false, a3, false, wb[li][3], (short)0, acc, false, false);
            acc = __builtin_amdgcn_wmma_f32_16x16x4_f32(false, a4, false, wb[li][4], (short)0, acc, false, false);

            h[li]    = acc;
            inpState = acc;
            asm volatile("" ::: "memory");  // next layer's LDS stores stay after this layer's loads
        }

        // Emit output = last layer's state (D layout -> [B, T, D]); pad lanes must not write
        if (n < DD) {
            #pragma unroll
            for (int j = 0; j < 8; ++j) {
                outLane[((size_t)j * TT + t) * DD] = h[LL - 1][j];
            }
        }
    }
}

extern "C" void kernel_launch(void* const* d_in, const int* in_sizes, int n_in,
                              void* d_out, int out_size, void* d_ws, size_t ws_size,
                              hipStream_t stream) {
    const float* initial = (const float*)d_in[0];
    const float* xseq    = (const float*)d_in[1];
    const float* Ws      = (const float*)d_in[2];
    const float* bs      = (const float*)d_in[3];
    float* out = (float*)d_out;

    dim3 grid(BB / 16);   // 256 single-wave workgroups, one 16-row batch tile each
    dim3 block(32);
    hipLaunchKernelGGL(rnn_wmma_kernel, grid, block, 0, stream,
                       initial, xseq, Ws, bs, out);
}